// Road2vec_75411035783382
// MI455X (gfx1250) — compile-verified
//
#include <hip/hip_runtime.h>
#include <stdint.h>

#define VOCAB 100000
#define EMB   128
#define EPS   1e-12f

typedef __attribute__((ext_vector_type(4))) float v4f;

// ---------------------------------------------------------------------------
// Kernel 1: tiled transpose  W[EMB][VOCAB] (row-major) -> Wt[VOCAB][EMB]
// VOCAB = 3125 * 32, EMB = 4 * 32  => perfect 32x32 tiling, no guards.
// ---------------------------------------------------------------------------
__global__ __launch_bounds__(256) void w_transpose(const float* __restrict__ W,
                                                   float* __restrict__ Wt) {
    __shared__ float tile[32][33];
    const int lx = threadIdx.x;          // 0..31 (lanes of a wave)
    const int ly = threadIdx.y;          // 0..7  (wave id in block)
    const int vTile = blockIdx.x * 32;
    const int eTile = blockIdx.y * 32;

#pragma unroll
    for (int i = 0; i < 4; ++i) {
        const int e = eTile + ly + 8 * i;
        tile[ly + 8 * i][lx] =
            __builtin_nontemporal_load(&W[(size_t)e * VOCAB + (size_t)(vTile + lx)]);
    }
    __syncthreads();
#pragma unroll
    for (int i = 0; i < 4; ++i) {
        const int v = vTile + ly + 8 * i;
        // regular (RT) store: Wt must stay L2-resident (51.2 MB << 192 MB L2)
        Wt[(size_t)v * EMB + (size_t)(eTile + lx)] = tile[lx][ly + 8 * i];
    }
}

// ---------------------------------------------------------------------------
// Kernel 2 (fast path): gather + bias + L2-normalize through async-to-LDS
// double buffering. One wave32 per token; 4 tokens per pipeline stage.
// Each global_load_async_to_lds_b128 moves one full 512 B embedding row
// (32 lanes x 16 B), tracked by ASYNCcnt; s_wait_asynccnt 0x4 retires the
// previous stage while the next stage's 4 rows are still in flight.
// ---------------------------------------------------------------------------
__global__ __launch_bounds__(256) void gather_norm_async(const int* __restrict__ x,
                                                         const float* __restrict__ Wt,
                                                         const float* __restrict__ bias,
                                                         float* __restrict__ out,
                                                         int ntok) {
    __shared__ __align__(16) float stage[2][8][4][EMB];   // 32 KB / block

    const int lane        = threadIdx.x & 31;
    const int waveInBlk   = threadIdx.x >> 5;
    const int wavesPerBlk = blockDim.x >> 5;
    const int gwave       = blockIdx.x * wavesPerBlk + waveInBlk;
    const int nwave       = gridDim.x * wavesPerBlk;

    const int nchunks = ntok >> 2;                        // 4 tokens per chunk
    const v4f bv = ((const v4f*)bias)[lane];              // b[4l .. 4l+3]

    auto issue = [&](int c, int buf) {
#pragma unroll
        for (int k = 0; k < 4; ++k) {
            const int v = x[4 * c + k];
            const float* src = Wt + (size_t)v * EMB + (size_t)lane * 4;
            // addr[31:0] of an LDS-aperture generic pointer == LDS byte offset
            uint32_t ldsAddr = (uint32_t)(uintptr_t)&stage[buf][waveInBlk][k][lane * 4];
            asm volatile("global_load_async_to_lds_b128 %0, %1, off"
                         :: "v"(ldsAddr), "v"(src)
                         : "memory");
        }
    };

    int buf = 0;
    int c   = gwave;
    if (c < nchunks) issue(c, 0);                         // prologue: stage 0

    for (; c < nchunks; c += nwave) {
        const int cn = c + nwave;
        if (cn < nchunks) {
            issue(cn, buf ^ 1);                           // next stage in flight
            asm volatile("s_wait_asynccnt 0x4" ::: "memory");  // prev 4 done
        } else {
            asm volatile("s_wait_asynccnt 0x0" ::: "memory");  // drain
        }

#pragma unroll
        for (int k = 0; k < 4; ++k) {
            v4f e = *(const v4f*)&stage[buf][waveInBlk][k][lane * 4];  // ds_load_b128
            e += bv;
            float ss = e.x * e.x + e.y * e.y + e.z * e.z + e.w * e.w;
#pragma unroll
            for (int m = 16; m >= 1; m >>= 1)
                ss += __shfl_xor(ss, m, 32);              // wave32 all-reduce
            const float inv = 1.0f / fmaxf(sqrtf(ss), EPS);
            v4f o = e * inv;
            // NT b128 store: 419 MB output stream must not evict Wt from L2
            __builtin_nontemporal_store(o, (v4f*)(out + (size_t)(4 * c + k) * EMB) + lane);
        }
        buf ^= 1;
    }

    // Tail (ntok % 4 != 0; unused for 819200 but kept for generality).
    if (gwave == 0) {
        for (int t = nchunks << 2; t < ntok; ++t) {
            const int v = x[t];
            v4f e = ((const v4f*)(Wt + (size_t)v * EMB))[lane];
            e += bv;
            float ss = e.x * e.x + e.y * e.y + e.z * e.z + e.w * e.w;
#pragma unroll
            for (int m = 16; m >= 1; m >>= 1)
                ss += __shfl_xor(ss, m, 32);
            const float inv = 1.0f / fmaxf(sqrtf(ss), EPS);
            v4f o = e * inv;
            __builtin_nontemporal_store(o, (v4f*)(out + (size_t)t * EMB) + lane);
        }
    }
}

// ---------------------------------------------------------------------------
// Kernel 2 (fallback): workspace too small for Wt -> strided column gather.
// ---------------------------------------------------------------------------
__global__ __launch_bounds__(256) void gather_norm_direct(const int* __restrict__ x,
                                                          const float* __restrict__ W,
                                                          const float* __restrict__ bias,
                                                          float* __restrict__ out,
                                                          int ntok) {
    const int lane        = threadIdx.x & 31;
    const int waveInBlk   = threadIdx.x >> 5;
    const int wavesPerBlk = blockDim.x >> 5;
    const int gwave       = blockIdx.x * wavesPerBlk + waveInBlk;
    const int nwave       = gridDim.x * wavesPerBlk;

    float bk[4];
#pragma unroll
    for (int k = 0; k < 4; ++k) bk[k] = bias[lane + 32 * k];

    for (int t = gwave; t < ntok; t += nwave) {
        const int v = x[t];
        const int tn = t + nwave;
        if (tn < ntok) {
            const int vn = x[tn];
            __builtin_prefetch(W + (size_t)lane * VOCAB + (size_t)vn, 0, 3);  // global_prefetch_b8
        }
        float e[4];
        float ss = 0.0f;
#pragma unroll
        for (int k = 0; k < 4; ++k) {
            e[k] = W[(size_t)(lane + 32 * k) * VOCAB + (size_t)v] + bk[k];
            ss += e[k] * e[k];
        }
#pragma unroll
        for (int m = 16; m >= 1; m >>= 1)
            ss += __shfl_xor(ss, m, 32);
        const float inv = 1.0f / fmaxf(sqrtf(ss), EPS);
#pragma unroll
        for (int k = 0; k < 4; ++k)
            __builtin_nontemporal_store(e[k] * inv, out + (size_t)t * EMB + 32 * k + lane);
    }
}

// ---------------------------------------------------------------------------
extern "C" void kernel_launch(void* const* d_in, const int* in_sizes, int n_in,
                              void* d_out, int out_size, void* d_ws, size_t ws_size,
                              hipStream_t stream) {
    const int*   x   = (const int*)d_in[0];    // [819200] int32 token ids
    const float* W   = (const float*)d_in[1];  // [EMB, VOCAB] fp32
    const float* b   = (const float*)d_in[2];  // [EMB] fp32
    float*       out = (float*)d_out;          // [ntok, EMB] fp32
    const int ntok = in_sizes[0];

    const size_t wtBytes = (size_t)VOCAB * EMB * sizeof(float);  // 51.2 MB

    if (ws_size >= wtBytes) {
        float* Wt = (float*)d_ws;
        dim3 tb(32, 8);
        dim3 tg(VOCAB / 32, EMB / 32);         // (3125, 4), exact
        w_transpose<<<tg, tb, 0, stream>>>(W, Wt);

        const int blocks = 2048;               // 16384 waves, ~12.5 chunks each
        gather_norm_async<<<blocks, 256, 0, stream>>>(x, Wt, b, out, ntok);
    } else {
        gather_norm_direct<<<2048, 256, 0, stream>>>(x, W, b, out, ntok);
    }
}